// MultiHeadAttentionRoPE_45629732553300
// MI455X (gfx1250) — compile-verified
//
#include <hip/hip_runtime.h>
#include <hip/hip_bf16.h>
#include <math.h>

typedef __attribute__((ext_vector_type(2))) float v2f;
typedef __attribute__((ext_vector_type(8))) float v8f;

#define NB 2
#define NS 2048
#define ND 1024
#define NH 16
#define NDH 64
#define N3D 3072

static __device__ __forceinline__ v8f wmma_f32(v2f a, v2f b, v8f c) {
    // V_WMMA_F32_16X16X4_F32 : D = A(16x4) * B(4x16) + C(16x16), fp32
    return __builtin_amdgcn_wmma_f32_16x16x4_f32(
        /*neg_a=*/false, a, /*neg_b=*/false, b,
        /*c_mod=*/(short)0, c, /*reuse_a=*/false, /*reuse_b=*/false);
}

// Async global -> LDS copies (GVS form: saddr base + per-lane vgpr offset).
// Tracked by ASYNCcnt.
static __device__ __forceinline__ void async_ld_b128(unsigned ldsByteOff,
                                                     unsigned long long base,
                                                     unsigned gByteOff) {
    asm volatile("global_load_async_to_lds_b128 %0, %1, %2"
                 :: "v"(ldsByteOff), "v"(gByteOff), "s"(base) : "memory");
}
static __device__ __forceinline__ void async_ld_b32(unsigned ldsByteOff,
                                                    unsigned long long base,
                                                    unsigned gByteOff) {
    asm volatile("global_load_async_to_lds_b32 %0, %1, %2"
                 :: "v"(ldsByteOff), "v"(gByteOff), "s"(base) : "memory");
}

// ---------------------------------------------------------------------------
// Kernel 1/4: C(M,N) = A(M,K) @ B(K,N), row-major fp32.
// Block: 256 threads (8 waves). Block tile 64x128, BK=16.
// Triple-buffered LDS filled by the async-DMA path; B stored pair-interleaved
// so each WMMA B-fragment is a single conflict-free ds_load_b64.
// Per k-block: preload 4 A-frags + 16 B-frags, then 16 back-to-back WMMAs.
// gridDim = (N/128, M/64)
// ---------------------------------------------------------------------------
#define BM 64
#define BN 128
#define BK 16
#define ASTRIDE 20   // padded A row stride (dwords): conflict-free, 16B-aligned rows

__global__ __launch_bounds__(256) void gemm_f32_wmma(
    const float* __restrict__ A, const float* __restrict__ B,
    float* __restrict__ C, int K, int N)
{
    __shared__ float As[3][BM * ASTRIDE];   // 3 x 5 KB
    __shared__ float Bs[3][(BK / 2) * BN * 2]; // 3 x 8 KB, pair-interleaved

    const int t    = threadIdx.x;
    const int lane = t & 31;
    const int wave = t >> 5;
    const int half = lane >> 4;
    const int ln   = lane & 15;
    const int wm   = wave & 3;     // 4 row strips of 16
    const int wn   = wave >> 2;    // 2 col strips of 64

    const int row0 = blockIdx.y * BM;
    const int col0 = blockIdx.x * BN;

    // cooperative-load mapping
    const int ar  = t >> 2;          // 0..63   A row in tile (b128)
    const int ak  = (t & 3) * 4;     // 0..12   A k chunk
    const int bkr = t >> 7;          // 0..1    B k parity (b32 x8)
    const int bcc = t & 127;         // 0..127  B column

    const unsigned long long Abase = (unsigned long long)(uintptr_t)A;
    const unsigned long long Bbase = (unsigned long long)(uintptr_t)B;

    const v8f zero8 = {0.f,0.f,0.f,0.f,0.f,0.f,0.f,0.f};
    v8f acc[4];
#pragma unroll
    for (int tt = 0; tt < 4; ++tt) acc[tt] = zero8;

    // 9 async ops per tile per wave (1 b128 + 8 b32)
    auto issue = [&](int kb, int buf) {
        unsigned aLds = (unsigned)(uintptr_t)&As[buf][ar * ASTRIDE + ak];
        unsigned aGo  = (unsigned)(((size_t)(row0 + ar) * K + kb * BK + ak) * 4);
        async_ld_b128(aLds, Abase, aGo);
#pragma unroll
        for (int i = 0; i < 8; ++i) {
            const int k = 2 * i + bkr;
            // pair-interleaved LDS: element B[k][c] -> dword (k>>1)*256 + c*2 + (k&1)
            unsigned bLds = (unsigned)(uintptr_t)&Bs[buf][i * 256 + bcc * 2 + bkr];
            unsigned bGo  = (unsigned)(((size_t)(kb * BK + k) * N + col0 + bcc) * 4);
            async_ld_b32(bLds, Bbase, bGo);
        }
    };

    auto compute = [&](int buf) {
        const float* as  = &As[buf][(16 * wm + ln) * ASTRIDE + 2 * half];
        const float* bsb = &Bs[buf][half * 256 + (64 * wn + ln) * 2];
        v2f af[4];
#pragma unroll
        for (int ks = 0; ks < 4; ++ks)
            af[ks] = *(const v2f*)(as + 4 * ks);          // ds_load_b64
        v2f bf[4][4];
#pragma unroll
        for (int ks = 0; ks < 4; ++ks)
#pragma unroll
            for (int tt = 0; tt < 4; ++tt)                 // ds_load_b64, no repack
                bf[ks][tt] = *(const v2f*)(bsb + ks * 512 + tt * 32);
#pragma unroll
        for (int ks = 0; ks < 4; ++ks)
#pragma unroll
            for (int tt = 0; tt < 4; ++tt)
                acc[tt] = wmma_f32(af[ks], bf[ks][tt], acc[tt]);
    };

    const int KB = K / BK;
    issue(0, 0);
    if (KB > 1) issue(1, 1);

    for (int kb = 0; kb < KB; ++kb) {
        // tile kb complete when at most the one newer tile (9 ops) is outstanding
        if (kb + 1 < KB) asm volatile("s_wait_asynccnt 0x9" ::: "memory");
        else             asm volatile("s_wait_asynccnt 0x0" ::: "memory");
        __syncthreads();
        if (kb + 2 < KB) issue(kb + 2, (kb + 2) % 3);  // into buffer last read at kb-1
        compute(kb % 3);
    }

    // C layout: VGPR v -> row (v + 8*half), lane ln -> column
#pragma unroll
    for (int tt = 0; tt < 4; ++tt) {
#pragma unroll
        for (int v = 0; v < 8; ++v) {
            C[(size_t)(row0 + 16 * wm + v + 8 * half) * N
              + col0 + 64 * wn + tt * 16 + ln] = acc[tt][v];
        }
    }
}

// ---------------------------------------------------------------------------
// Kernel 2: RoPE applied in place to the q and k slices of qkv (B,S,3D).
// ---------------------------------------------------------------------------
__global__ __launch_bounds__(256) void rope_inplace(float* __restrict__ qkv)
{
    int idx = blockIdx.x * blockDim.x + threadIdx.x;
    const int j = idx & 31;  idx >>= 5;
    const int h = idx & 15;  idx >>= 4;   // H = 16
    const int s = idx & 2047; idx >>= 11; // S = 2048
    const int b = idx;                    // B = 2

    const float inv_freq = __expf(-(float)j * (9.2103403719761836f / 32.0f));
    const float ang = (float)s * inv_freq;
    float c, sn;
    __sincosf(ang, &c, &sn);

    const size_t base = ((size_t)(b * NS + s)) * N3D + h * NDH;
#pragma unroll
    for (int off = 0; off <= ND; off += ND) {   // q at +0, k at +1024
        float x1 = qkv[base + off + j];
        float x2 = qkv[base + off + j + 32];
        qkv[base + off + j]      = x1 * c - x2 * sn;
        qkv[base + off + j + 32] = x2 * c + x1 * sn;
    }
}

// ---------------------------------------------------------------------------
// Kernel 3: causal flash attention. One wave per 16-query tile per (b,h).
// grid = (S/16, H, B), block = 32. K/V fragments preloaded so WMMA chains
// issue back-to-back.
// ---------------------------------------------------------------------------
__global__ __launch_bounds__(32) void flash_attn_wmma(
    const float* __restrict__ qkv, float* __restrict__ out)
{
    __shared__ float pLds[16 * 17];

    const int lane = threadIdx.x & 31;
    const int half = lane >> 4;
    const int ln   = lane & 15;

    const int q0 = blockIdx.x * 16;
    const int h  = blockIdx.y;
    const int b  = blockIdx.z;

    const float* Qb = qkv + (size_t)b * NS * N3D + h * NDH;
    const float* Kb = Qb + ND;
    const float* Vb = Qb + 2 * ND;

    v2f qf[16];
    {
        const float* qrow = Qb + (size_t)(q0 + ln) * N3D + 2 * half;
#pragma unroll
        for (int kk = 0; kk < 16; ++kk) {
            v2f tq = *(const v2f*)(qrow + 4 * kk);
            qf[kk] = tq * 0.125f;   // fold in 1/sqrt(Dh)
        }
    }

    float m[8], l[8];
#pragma unroll
    for (int v = 0; v < 8; ++v) { m[v] = -INFINITY; l[v] = 0.0f; }

    const v8f zero8 = {0.f,0.f,0.f,0.f,0.f,0.f,0.f,0.f};
    v8f o[4];
#pragma unroll
    for (int tt = 0; tt < 4; ++tt) o[tt] = zero8;

    const int ktMax = q0 >> 4;
    for (int kt = 0; kt <= ktMax; ++kt) {
        const int key0 = kt * 16;

        v2f kf[16];
        {
            const float* krow = Kb + (size_t)(key0 + ln) * N3D + 2 * half;
#pragma unroll
            for (int kk = 0; kk < 16; ++kk)
                kf[kk] = *(const v2f*)(krow + 4 * kk);
        }
        v2f vfr[4][4];
#pragma unroll
        for (int kk2 = 0; kk2 < 4; ++kk2) {
            const float* vr0 = Vb + (size_t)(key0 + 4 * kk2 + 2 * half) * N3D + ln;
#pragma unroll
            for (int tt = 0; tt < 4; ++tt) {
                vfr[kk2][tt][0] = vr0[tt * 16];
                vfr[kk2][tt][1] = vr0[tt * 16 + N3D];
            }
        }

        v8f sc = zero8;
#pragma unroll
        for (int kk = 0; kk < 16; ++kk)
            sc = wmma_f32(qf[kk], kf[kk], sc);

        if (kt == ktMax) {
#pragma unroll
            for (int v = 0; v < 8; ++v) {
                const int row = v + 8 * half;
                if (key0 + ln > q0 + row) sc[v] = -INFINITY;
            }
        }

#pragma unroll
        for (int v = 0; v < 8; ++v) {
            float red = sc[v];
            red = fmaxf(red, __shfl_xor(red, 1, 32));
            red = fmaxf(red, __shfl_xor(red, 2, 32));
            red = fmaxf(red, __shfl_xor(red, 4, 32));
            red = fmaxf(red, __shfl_xor(red, 8, 32));
            const float mn    = fmaxf(m[v], red);
            const float scale = __expf(m[v] - mn);
            const float p     = __expf(sc[v] - mn);
            float ps = p;
            ps += __shfl_xor(ps, 1, 32);
            ps += __shfl_xor(ps, 2, 32);
            ps += __shfl_xor(ps, 4, 32);
            ps += __shfl_xor(ps, 8, 32);
            l[v] = l[v] * scale + ps;
            m[v] = mn;
            sc[v] = p;
            o[0][v] *= scale; o[1][v] *= scale; o[2][v] *= scale; o[3][v] *= scale;
        }

        __syncthreads();
#pragma unroll
        for (int v = 0; v < 8; ++v)
            pLds[(v + 8 * half) * 17 + ln] = sc[v];
        __syncthreads();

#pragma unroll
        for (int kk2 = 0; kk2 < 4; ++kk2) {
            v2f pa;
            pa[0] = pLds[ln * 17 + 4 * kk2 + 2 * half];
            pa[1] = pLds[ln * 17 + 4 * kk2 + 2 * half + 1];
#pragma unroll
            for (int tt = 0; tt < 4; ++tt)
                o[tt] = wmma_f32(pa, vfr[kk2][tt], o[tt]);
        }
    }

#pragma unroll
    for (int v = 0; v < 8; ++v) {
        const float inv = 1.0f / l[v];
        const int row = v + 8 * half;
        float* orow = out + ((size_t)b * NS + q0 + row) * ND + h * NDH + ln;
#pragma unroll
        for (int tt = 0; tt < 4; ++tt)
            orow[tt * 16] = o[tt][v] * inv;
    }
}

// ---------------------------------------------------------------------------
extern "C" void kernel_launch(void* const* d_in, const int* in_sizes, int n_in,
                              void* d_out, int out_size, void* d_ws, size_t ws_size,
                              hipStream_t stream) {
    const float* x      = (const float*)d_in[0];   // (B,S,D)
    const float* w_qkv  = (const float*)d_in[1];   // (D,3D)
    const float* w_proj = (const float*)d_in[2];   // (D,D)
    float* out = (float*)d_out;                    // (B,S,D)

    float* qkv  = (float*)d_ws;                    // (B*S, 3D) = 48 MB
    float* attn = qkv + (size_t)NB * NS * N3D;     // (B*S, D)  = 16 MB

    // 1) qkv = x @ w_qkv
    gemm_f32_wmma<<<dim3(N3D / BN, (NB * NS) / BM), 256, 0, stream>>>(
        x, w_qkv, qkv, ND, N3D);

    // 2) RoPE in place on q,k slices
    {
        const int total = NB * NS * NH * 32;
        rope_inplace<<<total / 256, 256, 0, stream>>>(qkv);
    }

    // 3) causal flash attention -> attn (B,S,D)
    flash_attn_wmma<<<dim3(NS / 16, NH, NB), 32, 0, stream>>>(qkv, attn);

    // 4) out = attn @ w_proj
    gemm_f32_wmma<<<dim3(ND / BN, (NB * NS) / BM), 256, 0, stream>>>(
        attn, w_proj, out, ND, ND);
}